// LBPAttentionHead_90048284328256
// MI455X (gfx1250) — compile-verified
//
#include <hip/hip_runtime.h>
#include <stdint.h>

#define IMG_H 1024
#define IMG_W 1024
#define ROWS 8                  // output rows per workgroup band
#define HALO_ROWS (ROWS + 2)    // staged rows (1 halo row above/below)
#define LDS_STRIDE 1040         // 8 pad + 1024 data + 8 pad (keeps b128 16B-aligned)
#define DATA_OFF 8

typedef int v4i __attribute__((vector_size(16)));
typedef __attribute__((address_space(1))) v4i gv4i;   // global (device) space
typedef __attribute__((address_space(3))) v4i lv4i;   // LDS space

// ---- CDNA5 async global->LDS copy (ASYNCcnt-tracked) -----------------------
__device__ __forceinline__ void async_copy_b128(const float* gsrc, float* ldst) {
#if __has_builtin(__builtin_amdgcn_global_load_async_to_lds_b128)
    __builtin_amdgcn_global_load_async_to_lds_b128(
        (gv4i*)gsrc, (lv4i*)ldst, /*offset=*/0, /*cpol=*/0);
#else
    asm volatile("global_load_async_to_lds_b128 %0, %1, off"
                 :: "v"((unsigned)(uintptr_t)(lv4i*)ldst),
                    "v"((unsigned long long)(uintptr_t)gsrc)
                 : "memory");
#endif
}

__device__ __forceinline__ void wait_asynccnt0() {
#if __has_builtin(__builtin_amdgcn_s_wait_asynccnt)
    __builtin_amdgcn_s_wait_asynccnt(0);
#else
    asm volatile("s_wait_asynccnt 0" ::: "memory");
#endif
}

// floor(x*255) clipped to [0,255]  (uint8 quantization kept in f32)
__device__ __forceinline__ float quant(float v) {
    return fminf(fmaxf(floorf(v * 255.0f), 0.0f), 255.0f);
}

__global__ __launch_bounds__(256) void lbp_uniform_kernel(
        const float* __restrict__ x, float* __restrict__ out) {
    __shared__ float tile[HALO_ROWS][LDS_STRIDE];

    const int t    = threadIdx.x;     // 0..255
    const int band = blockIdx.x;      // 0..127
    const int chan = blockIdx.y;      // 0..31
    const int row0 = band * ROWS;     // first output row of this band
    const float* xc = x + (size_t)chan * IMG_H * IMG_W;

    // Zero the 8-float column pads of every halo row (160 stores total).
    if (t < HALO_ROWS * 16) {
        int hr  = t >> 4;
        int i   = t & 15;
        int col = (i < 8) ? i : (i - 8 + DATA_OFF + IMG_W);
        tile[hr][col] = 0.0f;
    }

    // Stage 10 halo rows: one async b128 per lane covers a full 4 KiB row.
#pragma unroll
    for (int hr = 0; hr < HALO_ROWS; ++hr) {
        int gr = row0 - 1 + hr;                 // uniform across the block
        if (gr >= 0 && gr < IMG_H) {
            async_copy_b128(xc + (size_t)gr * IMG_W + 4 * t,
                            &tile[hr][DATA_OFF + 4 * t]);
        } else {
            *(float4*)&tile[hr][DATA_OFF + 4 * t] = make_float4(0.f, 0.f, 0.f, 0.f);
        }
    }

    wait_asynccnt0();
    __syncthreads();

    // ---- compute: thread owns columns 4t..4t+3, rolling 3x6 register window
    const int cb = DATA_OFF + 4 * t;            // LDS col of first owned pixel

    // reference's rounded offsets: frac in {0.29289, 0.70711}
    const float f  = 0.29289f;
    const float F  = 0.70711f;
    const float wA = F * f;     // mixed weight
    const float wB = F * F;     // near weight
    const float wC = f * f;     // far weight
    const float inv255 = 1.0f / 255.0f;

    float buf[3][6];
#pragma unroll
    for (int j = 0; j < 6; ++j) buf[0][j] = quant(tile[0][cb - 1 + j]);
#pragma unroll
    for (int j = 0; j < 6; ++j) buf[1][j] = quant(tile[1][cb - 1 + j]);

#pragma unroll
    for (int r = 0; r < ROWS; ++r) {
#pragma unroll
        for (int j = 0; j < 6; ++j)
            buf[(r + 2) % 3][j] = quant(tile[r + 2][cb - 1 + j]);

        const float* a = buf[r % 3];            // row r-1
        const float* b = buf[(r + 1) % 3];      // row r
        const float* c = buf[(r + 2) % 3];      // row r+1

        float4 o;
        float* op = &o.x;
#pragma unroll
        for (int j = 0; j < 4; ++j) {
            float nw = a[j], n  = a[j + 1], ne = a[j + 2];
            float w_ = b[j], ct = b[j + 1], e  = b[j + 2];
            float sw = c[j], s  = c[j + 1], se = c[j + 2];

            // bilinear diagonal samples (contraction allowed -> mul + 3 fma each)
            float v1 = wA * n  + wB * ne + wC * ct + wA * e;   // k=1 (NE)
            float v3 = wB * nw + wA * n  + wA * w_ + wC * ct;  // k=3 (NW)
            float v5 = wA * w_ + wC * ct + wB * sw + wA * s;   // k=5 (SW)
            float v7 = wC * ct + wA * e  + wA * s  + wB * se;  // k=7 (SE)

            unsigned m = 0;
            m |= (e  >= ct) ? 1u   : 0u;   // k=0
            m |= (v1 >= ct) ? 2u   : 0u;   // k=1
            m |= (n  >= ct) ? 4u   : 0u;   // k=2
            m |= (v3 >= ct) ? 8u   : 0u;   // k=3
            m |= (w_ >= ct) ? 16u  : 0u;   // k=4
            m |= (v5 >= ct) ? 32u  : 0u;   // k=5
            m |= (s  >= ct) ? 64u  : 0u;   // k=6
            m |= (v7 >= ct) ? 128u : 0u;   // k=7

            unsigned rot = ((m << 1) | (m >> 7)) & 0xFFu;
            int trans = __popc(m ^ rot);
            int ones  = __popc(m);
            float lbp = (trans <= 2) ? (float)ones : 9.0f;
            op[j] = lbp * inv255;
        }

        *(float4*)&out[((size_t)chan * IMG_H + row0 + r) * IMG_W + 4 * t] = o;
    }
}

extern "C" void kernel_launch(void* const* d_in, const int* in_sizes, int n_in,
                              void* d_out, int out_size, void* d_ws, size_t ws_size,
                              hipStream_t stream) {
    (void)n_in; (void)d_ws; (void)ws_size; (void)out_size;
    const float* x = (const float*)d_in[0];
    float* out = (float*)d_out;
    int channels = in_sizes[0] / (IMG_H * IMG_W);   // 32
    dim3 grid(IMG_H / ROWS, channels, 1);           // (128, 32)
    dim3 block(256, 1, 1);
    lbp_uniform_kernel<<<grid, block, 0, stream>>>(x, out);
}